// TRIX_69398081568799
// MI455X (gfx1250) — compile-verified
//
#include <hip/hip_runtime.h>
#include <cstddef>

// ---------------------------------------------------------------------------
// NBFNet-style relational GNN forward for MI455X (gfx1250, wave32).
// Dense matmuls: V_WMMA_F32_16X16X4_F32, weights staged in LDS (transposed,
// padded stride -> conflict-free ds_load_b64), each wave computes a 16x64
// output strip so every A global load feeds 4 WMMAs. Row guard is done by
// address clamping (no divergent branch in the k loop).
// Edge aggregation: global_atomic_add_f32. LayerNorm: wave32 shuffles.
// ---------------------------------------------------------------------------

#define Bc   4
#define NEGc 33
#define Nn   10000
#define Rr   256
#define Dd   64
#define Ll   6
#define Ee   200000
#define ERe  80000
#define Ff   128

#define KMAX 128
#define KPAD (KMAX + 4)   // 132: bank-stride 4 -> all 32 lanes hit distinct banks

typedef __attribute__((ext_vector_type(2))) float v2f;
typedef __attribute__((ext_vector_type(8))) float v8f;

// ---------------------------------------------------------------------------
// WMMA GEMM:  Y[rows x Dout] = act( [A1|A2][rows x (K1+K2)] @ W + bias )
// A2 == nullptr -> plain GEMM over K1. W is (K1+K2) x Dout row-major.
// Dout must be a multiple of 64 (here: 64 or 128); K1,K2 multiples of 4.
// Block = 8 waves; block owns one 64-column group (weights in LDS), each wave
// owns one 16-row tile and produces 16x64 via 4 accumulators.
// ---------------------------------------------------------------------------
__global__ __launch_bounds__(256) void k_gemm_wmma(
    const float* __restrict__ A1, const float* __restrict__ A2,
    const float* __restrict__ W, const float* __restrict__ bias,
    float* __restrict__ Y, int rows, int K1, int K2, int Dout, int doRelu) {
  __shared__ float ldsW[64 * KPAD];
  __shared__ float ldsB[64];

  const int Ktot  = K1 + K2;
  const int nCG   = Dout >> 6;              // 64-wide column groups
  const int cg    = blockIdx.x % nCG;
  const int col0g = cg << 6;

  // Stage W^T panel: ldsW[c*KPAD + k] = W[k*Dout + col0g + c] (coalesced read)
  for (int idx = threadIdx.x; idx < (Ktot << 6); idx += 256) {
    const int k = idx >> 6;
    const int c = idx & 63;
    ldsW[c * KPAD + k] = W[(size_t)k * Dout + col0g + c];
  }
  if (threadIdx.x < 64) ldsB[threadIdx.x] = bias ? bias[col0g + threadIdx.x] : 0.f;
  __syncthreads();

  const int lane = threadIdx.x & 31;
  const int half = lane >> 4;               // 0: K pair (k,k+1)  1: (k+2,k+3)
  const int l16  = lane & 15;               // A: row-in-tile, B/C/D: col-in-tile
  const int wiB  = threadIdx.x >> 5;        // wave in block, 0..7
  const int nRT  = (rows + 15) >> 4;        // 16-row tiles
  const int nRG  = (nRT + 7) >> 3;          // row groups of 8 tiles
  const int gRG  = gridDim.x / nCG;         // row-group stride over grid
  const int h2   = half << 1;

  for (int rg = blockIdx.x / nCG; rg < nRG; rg += gRG) {
    const int rt = (rg << 3) + wiB;
    if (rt >= nRT) continue;
    const int row0 = rt << 4;
    const int row  = row0 + l16;
    // Clamp instead of guard: out-of-range rows read a valid address; their
    // accumulator rows are never stored (row m of D depends only on row m of A).
    const int rowC = (row < rows) ? row : (rows - 1);
    const float* pa1 = A1 + (size_t)rowC * K1 + h2;
    const float* pa2 = A2 ? (A2 + (size_t)rowC * K2 + h2) : nullptr;
    const float* lw  = &ldsW[l16 * KPAD + h2];

    v8f acc[4];
    acc[0] = (v8f){}; acc[1] = (v8f){}; acc[2] = (v8f){}; acc[3] = (v8f){};

    for (int k = 0; k < K1; k += 4) {
      const v2f a  = *(const v2f*)(pa1 + k);
      const v2f b0 = *(const v2f*)(lw + k);
      const v2f b1 = *(const v2f*)(lw + 16 * KPAD + k);
      const v2f b2 = *(const v2f*)(lw + 32 * KPAD + k);
      const v2f b3 = *(const v2f*)(lw + 48 * KPAD + k);
      acc[0] = __builtin_amdgcn_wmma_f32_16x16x4_f32(false, a, false, b0, (short)0, acc[0], false, false);
      acc[1] = __builtin_amdgcn_wmma_f32_16x16x4_f32(false, a, false, b1, (short)0, acc[1], false, false);
      acc[2] = __builtin_amdgcn_wmma_f32_16x16x4_f32(false, a, false, b2, (short)0, acc[2], false, false);
      acc[3] = __builtin_amdgcn_wmma_f32_16x16x4_f32(false, a, false, b3, (short)0, acc[3], false, false);
    }
    for (int k = 0; k < K2; k += 4) {
      const v2f a  = *(const v2f*)(pa2 + k);
      const v2f b0 = *(const v2f*)(lw + (K1 + k));
      const v2f b1 = *(const v2f*)(lw + 16 * KPAD + (K1 + k));
      const v2f b2 = *(const v2f*)(lw + 32 * KPAD + (K1 + k));
      const v2f b3 = *(const v2f*)(lw + 48 * KPAD + (K1 + k));
      acc[0] = __builtin_amdgcn_wmma_f32_16x16x4_f32(false, a, false, b0, (short)0, acc[0], false, false);
      acc[1] = __builtin_amdgcn_wmma_f32_16x16x4_f32(false, a, false, b1, (short)0, acc[1], false, false);
      acc[2] = __builtin_amdgcn_wmma_f32_16x16x4_f32(false, a, false, b2, (short)0, acc[2], false, false);
      acc[3] = __builtin_amdgcn_wmma_f32_16x16x4_f32(false, a, false, b3, (short)0, acc[3], false, false);
    }

    const int rbase = row0 + (half << 3);
    if (row0 + 16 <= rows) {
      // Full tile: unconditional stores
#pragma unroll
      for (int ct = 0; ct < 4; ++ct) {
        const int col  = col0g + (ct << 4) + l16;
        const float bv = ldsB[(ct << 4) + l16];
        float* yp = Y + (size_t)rbase * Dout + col;
#pragma unroll
        for (int i2 = 0; i2 < 8; ++i2) {
          float v = acc[ct][i2] + bv;
          if (doRelu) v = fmaxf(v, 0.f);
          yp[(size_t)i2 * Dout] = v;
        }
      }
    } else {
#pragma unroll
      for (int ct = 0; ct < 4; ++ct) {
        const int col  = col0g + (ct << 4) + l16;
        const float bv = ldsB[(ct << 4) + l16];
#pragma unroll
        for (int i2 = 0; i2 < 8; ++i2) {
          const int r = rbase + i2;
          if (r < rows) {
            float v = acc[ct][i2] + bv;
            if (doRelu) v = fmaxf(v, 0.f);
            Y[(size_t)r * Dout + col] = v;
          }
        }
      }
    }
  }
}

// ---------------------------------------------------------------------------
// Edge message + scatter:  agg[b,dst[e],:] += x[b,src[e],:] * rel[b,et[e],:]
// 16 lanes (float4 each) cover an edge's 64 features. fp32 global atomics.
// ---------------------------------------------------------------------------
__global__ void k_edge(const float* __restrict__ x, const float* __restrict__ rel,
                       float* __restrict__ agg,
                       const int* __restrict__ src, const int* __restrict__ dst,
                       const int* __restrict__ et,
                       int nE, int xRows, int relRows) {
  const size_t total  = (size_t)nE * 64;            // nE * B(4) * 16 chunks
  const size_t stride = (size_t)gridDim.x * blockDim.x;
  for (size_t i = (size_t)blockIdx.x * blockDim.x + threadIdx.x; i < total; i += stride) {
    const int d4 = (int)(i & 15);
    const int b  = (int)((i >> 4) & 3);
    const int e  = (int)(i >> 6);
    const int s  = src[e];
    const int d  = dst[e];
    const int tt = et[e];
    const float4 xv = *(const float4*)(x   + (((size_t)b * xRows + s)   << 6) + d4 * 4);
    const float4 rv = *(const float4*)(rel + (((size_t)b * relRows + tt) << 6) + d4 * 4);
    float* ap = agg + (((size_t)b * xRows + d) << 6) + d4 * 4;
    unsafeAtomicAdd(ap + 0, xv.x * rv.x);
    unsafeAtomicAdd(ap + 1, xv.y * rv.y);
    unsafeAtomicAdd(ap + 2, xv.z * rv.z);
    unsafeAtomicAdd(ap + 3, xv.w * rv.w);
  }
}

// ---------------------------------------------------------------------------
// dst[row,:] += relu(layernorm(src[row,:]) * gam + bet), D=64, 1 wave / row.
// ---------------------------------------------------------------------------
__global__ void k_ln_relu_accum(const float* __restrict__ src,
                                const float* __restrict__ gam,
                                const float* __restrict__ bet,
                                float* __restrict__ dst, int rows) {
  const int wid  = (int)((blockIdx.x * blockDim.x + threadIdx.x) >> 5);
  const int nW   = (int)((gridDim.x * blockDim.x) >> 5);
  const int lane = threadIdx.x & 31;
  for (int r = wid; r < rows; r += nW) {
    const float* p = src + (size_t)r * 64;
    const float v0 = p[lane];
    const float v1 = p[lane + 32];
    float s = v0 + v1;
    for (int off = 16; off; off >>= 1) s += __shfl_xor(s, off, 32);
    const float mean = s * (1.f / 64.f);
    const float d0 = v0 - mean, d1 = v1 - mean;
    float sq = d0 * d0 + d1 * d1;
    for (int off = 16; off; off >>= 1) sq += __shfl_xor(sq, off, 32);
    const float rstd = rsqrtf(sq * (1.f / 64.f) + 1e-5f);
    const float y0 = fmaxf(d0 * rstd * gam[lane]      + bet[lane],      0.f);
    const float y1 = fmaxf(d1 * rstd * gam[lane + 32] + bet[lane + 32], 0.f);
    float* q = dst + (size_t)r * 64;
    q[lane]      += y0;
    q[lane + 32] += y1;
  }
}

// ----------------------------- small helpers -------------------------------
__global__ void k_fill(float* p, size_t n, float v) {
  size_t st = (size_t)gridDim.x * blockDim.x;
  for (size_t i = (size_t)blockIdx.x * blockDim.x + threadIdx.x; i < n; i += st) p[i] = v;
}

__global__ void k_add(float* __restrict__ dst, const float* __restrict__ src, size_t n) {
  size_t st = (size_t)gridDim.x * blockDim.x;
  for (size_t i = (size_t)blockIdx.x * blockDim.x + threadIdx.x; i < n; i += st)
    dst[i] += src[i];
}

// boundary[b, idx(b), :] += q[b,:]   idx from batch column `col` (0=h0, 2=r0)
__global__ void k_boundary_add(float* __restrict__ bnd, const float* __restrict__ q,
                               const int* __restrict__ batch, int col, int rowsPer) {
  int tid = blockIdx.x * blockDim.x + threadIdx.x;
  if (tid >= Bc * Dd) return;
  const int b = tid >> 6, d = tid & 63;
  const int idx = batch[b * NEGc * 3 + col];
  bnd[((size_t)b * rowsPer + idx) * 64 + d] += q[b * 64 + d];
}

// q[b,:] = relcur[b, r0[b], :]
__global__ void k_gather_q(const float* __restrict__ relcur,
                           const int* __restrict__ batch, float* __restrict__ q) {
  int tid = blockIdx.x * blockDim.x + threadIdx.x;
  if (tid >= Bc * Dd) return;
  const int b = tid >> 6, d = tid & 63;
  const int r0 = batch[b * NEGc * 3 + 2];
  q[tid] = relcur[((size_t)b * Rr + r0) * 64 + d];
}

// feat[b,n,:] = concat(x[b,n,:], q[b,:])
__global__ void k_build_feat(const float* __restrict__ x, const float* __restrict__ q,
                             float* __restrict__ feat) {
  const size_t n  = (size_t)Bc * Nn * Ff;
  const size_t st = (size_t)gridDim.x * blockDim.x;
  for (size_t i = (size_t)blockIdx.x * blockDim.x + threadIdx.x; i < n; i += st) {
    const int    f  = (int)(i & 127);
    const size_t bn = i >> 7;               // b*N + n
    const int    b  = (int)(bn / Nn);
    feat[i] = (f < 64) ? x[bn * 64 + f] : q[b * 64 + (f - 64)];
  }
}

// g[b,j,:] = feat[b, t_idx[b,j], :]
__global__ void k_gather_t(const float* __restrict__ feat,
                           const int* __restrict__ batch, float* __restrict__ g) {
  const int n  = Bc * NEGc * Ff;
  const int st = gridDim.x * blockDim.x;
  for (int i = blockIdx.x * blockDim.x + threadIdx.x; i < n; i += st) {
    const int f  = i & 127;
    const int bj = i >> 7;                  // b*NEG + j
    const int b  = bj / NEGc;
    const int j  = bj - b * NEGc;
    const int t  = batch[(b * NEGc + j) * 3 + 1];
    g[i] = feat[((size_t)b * Nn + t) * Ff + f];
  }
}

// out[r] = dot(hid[r,:128], w2[:,0]) + b2[0], one wave per row
__global__ void k_score(const float* __restrict__ hid, const float* __restrict__ w2,
                        const float* __restrict__ b2, float* __restrict__ out, int rows) {
  const int wid  = (int)((blockIdx.x * blockDim.x + threadIdx.x) >> 5);
  const int nW   = (int)((gridDim.x * blockDim.x) >> 5);
  const int lane = threadIdx.x & 31;
  for (int r = wid; r < rows; r += nW) {
    const float* h = hid + (size_t)r * 128;
    float s = 0.f;
    for (int k = lane; k < 128; k += 32) s += h[k] * w2[k];
    for (int off = 16; off; off >>= 1) s += __shfl_xor(s, off, 32);
    if (lane == 0) out[r] = s + b2[0];
  }
}

// ---------------------------------------------------------------------------
extern "C" void kernel_launch(void* const* d_in, const int* in_sizes, int n_in,
                              void* d_out, int out_size, void* d_ws, size_t ws_size,
                              hipStream_t stream) {
  (void)in_sizes; (void)n_in; (void)out_size; (void)ws_size;

  // --- inputs in setup_inputs() dict order ---
  const int*   edge_src  = (const int*)d_in[0];
  const int*   edge_dst  = (const int*)d_in[1];
  const int*   edge_type = (const int*)d_in[2];
  const int*   rel_src   = (const int*)d_in[3];
  const int*   rel_dst   = (const int*)d_in[4];
  const int*   rel_etype = (const int*)d_in[5];
  const int*   batch     = (const int*)d_in[6];
  const float* e1p1w = (const float*)d_in[7];
  const float* e1p1b = (const float*)d_in[8];
  const float* e1p2w = (const float*)d_in[9];
  const float* e1p2b = (const float*)d_in[10];
  const float* e1lw  = (const float*)d_in[11];
  const float* e1lb  = (const float*)d_in[12];
  const float* e1lns = (const float*)d_in[13];
  const float* e1lnb = (const float*)d_in[14];
  const float* e2p1w = (const float*)d_in[15];
  const float* e2p1b = (const float*)d_in[16];
  const float* e2p2w = (const float*)d_in[17];
  const float* e2p2b = (const float*)d_in[18];
  const float* e2lw  = (const float*)d_in[19];
  const float* e2lb  = (const float*)d_in[20];
  const float* e2lns = (const float*)d_in[21];
  const float* e2lnb = (const float*)d_in[22];
  const float* e2m1w = (const float*)d_in[23];
  const float* e2m1b = (const float*)d_in[24];
  const float* e2m2w = (const float*)d_in[25];
  const float* e2m2b = (const float*)d_in[26];
  const float* rp1w  = (const float*)d_in[27];
  const float* rp1b  = (const float*)d_in[28];
  const float* rp2w  = (const float*)d_in[29];
  const float* rp2b  = (const float*)d_in[30];
  const float* rlw   = (const float*)d_in[31];
  const float* rlb   = (const float*)d_in[32];
  const float* rlns  = (const float*)d_in[33];
  const float* rlnb  = (const float*)d_in[34];
  const float* nmw   = (const float*)d_in[35];
  const float* nmb   = (const float*)d_in[36];

  // --- workspace carve (~74 MB of f32) ---
  const size_t BND = (size_t)Bc * Nn * Dd;   // 2,560,000
  const size_t BRD = (size_t)Bc * Rr * Dd;   //    65,536
  const size_t BNF = (size_t)Bc * Nn * Ff;   // 5,120,000
  float* wsf = (float*)d_ws;
  size_t o = 0;
  float* node_reps = wsf + o; o += BND;
  float* hidN  = wsf + o; o += BND;   // rel-graph MLP hidden; aliased as x_e
  float* relN  = wsf + o; o += BND;   // rel-graph MLP out;   aliased as boundary_e
  float* agg_e = wsf + o; o += BND;
  float* out_e = wsf + o; o += BND;
  float* feat  = wsf + o; o += BNF;
  float* xr    = wsf + o; o += BRD;
  float* bnd_r = wsf + o; o += BRD;
  float* h_r   = wsf + o; o += BRD;
  float* agg_r = wsf + o; o += BRD;
  float* out_r = wsf + o; o += BRD;
  float* hidR  = wsf + o; o += BRD;
  float* relE  = wsf + o; o += BRD;
  float* qbuf  = wsf + o; o += (size_t)Bc * Dd;
  float* gbuf  = wsf + o; o += (size_t)Bc * NEGc * Ff;
  float* hid_s = wsf + o; o += (size_t)Bc * NEGc * Ff;

  auto eb = [](size_t n) -> int {
    size_t b = (n + 255) / 256;
    return (int)(b > 16384 ? 16384 : b);
  };
  auto gemm = [&](const float* A1, const float* A2, const float* Wt, const float* bias,
                  float* Y, int rows, int K1, int K2, int Dout, int relu) {
    const int nRT = (rows + 15) >> 4;
    const int nRG = (nRT + 7) >> 3;          // 8 row-tiles (128 rows) per block
    const int nCG = Dout >> 6;               // 64-wide column groups
    int rgLaunch = nRG > 2048 ? 2048 : nRG;
    int blocks = rgLaunch * nCG;             // gridDim multiple of nCG
    k_gemm_wmma<<<blocks, 256, 0, stream>>>(A1, A2, Wt, bias, Y, rows, K1, K2, Dout, relu);
  };

  // entity Bellman-Ford pass (6 layers) over the entity graph -> feat (B,N,F)
  auto entity = [&](const float* relcur,
                    const float* p1w, const float* p1b, const float* p2w, const float* p2b,
                    const float* lw, const float* lb, const float* lns, const float* lnb) {
    float* x_e   = hidN;   // reuse rel-graph scratch
    float* bnd_e = relN;
    k_gather_q<<<1, 256, 0, stream>>>(relcur, batch, qbuf);
    k_fill<<<eb(BND), 256, 0, stream>>>(bnd_e, BND, 0.f);
    k_boundary_add<<<1, 256, 0, stream>>>(bnd_e, qbuf, batch, /*col=h0*/0, Nn);
    hipMemcpyAsync(x_e, bnd_e, BND * sizeof(float), hipMemcpyDeviceToDevice, stream);
    for (int i = 0; i < Ll; ++i) {
      gemm(relcur, nullptr, p1w + (size_t)i * Dd * Dd, p1b + (size_t)i * Dd,
           hidR, Bc * Rr, Dd, 0, Dd, 1);
      gemm(hidR, nullptr, p2w + (size_t)i * Dd * Dd, p2b + (size_t)i * Dd,
           relE, Bc * Rr, Dd, 0, Dd, 0);
      hipMemcpyAsync(agg_e, bnd_e, BND * sizeof(float), hipMemcpyDeviceToDevice, stream);
      k_edge<<<16384, 256, 0, stream>>>(x_e, relE, agg_e,
                                        edge_src, edge_dst, edge_type, Ee, Nn, Rr);
      gemm(x_e, agg_e, lw + (size_t)i * Ff * Dd, lb + (size_t)i * Dd,
           out_e, Bc * Nn, Dd, Dd, Dd, 0);
      k_ln_relu_accum<<<(Bc * Nn + 7) / 8, 256, 0, stream>>>(
          out_e, lns + (size_t)i * Dd, lnb + (size_t)i * Dd, x_e, Bc * Nn);
    }
    k_build_feat<<<eb(BNF), 256, 0, stream>>>(x_e, qbuf, feat);
  };

  // --- relation-graph init: node_reps = 1, boundary_r[b, r0[b], :] = 1 ---
  k_fill<<<eb(BND), 256, 0, stream>>>(node_reps, BND, 1.f);
  k_fill<<<1, 256, 0, stream>>>(qbuf, (size_t)Bc * Dd, 1.f);
  k_fill<<<eb(BRD), 256, 0, stream>>>(bnd_r, BRD, 0.f);
  k_boundary_add<<<1, 256, 0, stream>>>(bnd_r, qbuf, batch, /*col=r0*/2, Rr);
  hipMemcpyAsync(xr, bnd_r, BRD * sizeof(float), hipMemcpyDeviceToDevice, stream);

  // --- relation-graph Bellman-Ford: 6 layers x 4 edge groups ---
  for (int i = 0; i < Ll; ++i) {
    k_fill<<<eb(BRD), 256, 0, stream>>>(h_r, BRD, 0.f);
    for (int g = 0; g < 4; ++g) {
      const size_t wo = (size_t)g * Ll + i;
      gemm(node_reps, nullptr, rp1w + wo * Dd * Dd, rp1b + wo * Dd,
           hidN, Bc * Nn, Dd, 0, Dd, 1);
      gemm(hidN, nullptr, rp2w + wo * Dd * Dd, rp2b + wo * Dd,
           relN, Bc * Nn, Dd, 0, Dd, 0);
      hipMemcpyAsync(agg_r, bnd_r, BRD * sizeof(float), hipMemcpyDeviceToDevice, stream);
      k_edge<<<4096, 256, 0, stream>>>(xr, relN, agg_r,
                                       rel_src + (size_t)g * ERe,
                                       rel_dst + (size_t)g * ERe,
                                       rel_etype + (size_t)g * ERe, ERe, Rr, Nn);
      gemm(xr, agg_r, rlw + wo * Ff * Dd, rlb + wo * Dd, out_r, Bc * Rr, Dd, Dd, Dd, 0);
      k_ln_relu_accum<<<(Bc * Rr + 7) / 8, 256, 0, stream>>>(
          out_r, rlns + wo * Dd, rlnb + wo * Dd, h_r, Bc * Rr);
    }
    k_add<<<eb(BRD), 256, 0, stream>>>(xr, h_r, BRD);
    if (i == 0) {
      entity(xr, e1p1w, e1p1b, e1p2w, e1p2b, e1lw, e1lb, e1lns, e1lnb);
      gemm(feat, nullptr, nmw, nmb, node_reps, Bc * Nn, Ff, 0, Dd, 0);
    }
  }

  // --- second entity pass with final rel_reps, then scoring MLP ---
  entity(xr, e2p1w, e2p1b, e2p2w, e2p2b, e2lw, e2lb, e2lns, e2lnb);
  k_gather_t<<<eb((size_t)Bc * NEGc * Ff), 256, 0, stream>>>(feat, batch, gbuf);
  gemm(gbuf, nullptr, e2m1w, e2m1b, hid_s, Bc * NEGc, Ff, 0, Ff, 1);
  k_score<<<(Bc * NEGc * 32 + 255) / 256, 256, 0, stream>>>(
      hid_s, e2m2w, e2m2b, (float*)d_out, Bc * NEGc);
}